// LocalConvolution_26439818674593
// MI455X (gfx1250) — compile-verified
//
#include <hip/hip_runtime.h>

// Problem constants (from the reference)
#define KSZ   5
#define PADC  2
#define NB    4
#define CC    64     // input channels
#define CWX   8      // weight channels (c % 8 selects weight channel)
#define HH    128
#define WW    128
#define HOXWO (HH * WW)

// Tiling
#define TH    8
#define TW    32
#define TROW  (TH + 4)              // 12 rows incl. halo
#define ROWD  40                    // padded LDS row stride (dwords)
#define COL0  2                     // tile col 0 lives at dword 2 of each row
                                    // -> interior cols (2+4q) land 16B-aligned
#define CH_STRIDE  (TROW * ROWD)    // 480 dwords per channel
#define TILE_ELEMS (CC * CH_STRIDE) // 30720 floats
#define TILE_BYTES (TILE_ELEMS * 4) // 122880 bytes of LDS (2 WGs / 320KB WGP)

__global__ __launch_bounds__(256)
void localconv_gfx1250_kernel(const float* __restrict__ in,
                              const float* __restrict__ wgt,
                              float* __restrict__ out) {
    extern __shared__ float tile[];   // [CC][TROW][ROWD], cols at dw COL0..COL0+35
    const int tid = threadIdx.x;      // 0..255
    const int bx  = blockIdx.x;       // 0..3   (W tiles)
    const int by  = blockIdx.y;       // 0..15  (H tiles)
    const int n   = blockIdx.z;       // 0..3

    // ---- 1) zero LDS tile (halo beyond image must read as zero) ----
    {
        float4 z4 = make_float4(0.f, 0.f, 0.f, 0.f);
        float4* t4 = (float4*)tile;
        #pragma unroll
        for (int q = 0; q < TILE_ELEMS / 4 / 256; ++q)   // 30 iterations
            t4[tid + q * 256] = z4;
    }
    __syncthreads();

    const int gy0 = by * TH - PADC;
    const int gx0 = bx * TW - PADC;

    // ---- 2a) async global->LDS, interior columns as B128 ----
    // interior cols col = 2+4q, q=0..7: gx = bx*32 + 4q, always in-range,
    // 16B-aligned in global AND in LDS (row stride 40 dw, col0 at dw 2).
    for (int idx = tid; idx < CC * TROW * 8; idx += 256) {   // 24 iters
        const int c   = idx / (TROW * 8);
        const int rem = idx - c * (TROW * 8);
        const int r   = rem >> 3;
        const int q   = rem & 7;
        const int gy  = gy0 + r;
        if ((unsigned)gy < (unsigned)HH) {
            const unsigned lds_off =
                (unsigned)(c * CH_STRIDE + r * ROWD + COL0 + 2 + 4 * q) * 4u;
            const unsigned goff =
                (unsigned)(((n * CC + c) * HH + gy) * WW + bx * TW + 4 * q) * 4u;
            asm volatile("global_load_async_to_lds_b128 %0, %1, %2"
                         :
                         : "v"(lds_off), "v"(goff), "s"(in)
                         : "memory");
        }
    }

    // ---- 2b) async global->LDS, halo columns {0,1,34,35} as B32 ----
    for (int idx = tid; idx < CC * TROW * 4; idx += 256) {   // 12 iters
        const int c   = idx / (TROW * 4);
        const int rem = idx - c * (TROW * 4);
        const int r   = rem >> 2;
        const int e   = rem & 3;
        const int col = (e < 2) ? e : (32 + e);   // 0,1,34,35
        const int gy  = gy0 + r;
        const int gx  = gx0 + col;
        if ((unsigned)gy < (unsigned)HH && (unsigned)gx < (unsigned)WW) {
            const unsigned lds_off =
                (unsigned)(c * CH_STRIDE + r * ROWD + COL0 + col) * 4u;
            const unsigned goff =
                (unsigned)(((n * CC + c) * HH + gy) * WW + gx) * 4u;
            asm volatile("global_load_async_to_lds_b32 %0, %1, %2"
                         :
                         : "v"(lds_off), "v"(goff), "s"(in)
                         : "memory");
        }
    }

    asm volatile("s_wait_asynccnt 0" ::: "memory");
    __syncthreads();

    // ---- 3) compute: one output pixel per thread, all 64 channels ----
    const int px = tid & 31;          // lane dim -> contiguous wo (coalesced)
    const int py = tid >> 5;          // 0..7
    const int ho = by * TH + py;
    const int wo = bx * TW + px;
    const int p  = ho * WW + wo;      // pixel index in Ho*Wo

    float acc[CC];
    #pragma unroll
    for (int c = 0; c < CC; ++c) acc[c] = 0.f;

    // weight base for this (n, pixel): layout [N][CW][K*K][Ho*Wo]
    const float* wb = wgt + (size_t)n * CWX * KSZ * KSZ * HOXWO + p;

    #pragma unroll 1
    for (int i = 0; i < KSZ; ++i) {
        #pragma unroll 1
        for (int j = 0; j < KSZ; ++j) {
            const int kk = i * KSZ + j;
            // 8 per-pixel weights for this tap; streamed non-temporally
            // (52 MB tensor, read exactly once), reused across 8 groups.
            float wv[CWX];
            #pragma unroll
            for (int wc = 0; wc < CWX; ++wc)
                wv[wc] = __builtin_nontemporal_load(
                    wb + (size_t)(wc * KSZ * KSZ + kk) * HOXWO);
            const int lb = (py + i) * ROWD + (px + j) + COL0;
            #pragma unroll
            for (int c = 0; c < CC; ++c)
                acc[c] += tile[c * CH_STRIDE + lb] * wv[c & 7];
        }
    }

    // ---- 4) store: coalesced, non-temporal (no reuse) ----
    float* ob = out + (size_t)n * CC * HOXWO + p;
    #pragma unroll
    for (int c = 0; c < CC; ++c)
        __builtin_nontemporal_store(acc[c], ob + (size_t)c * HOXWO);
}

extern "C" void kernel_launch(void* const* d_in, const int* in_sizes, int n_in,
                              void* d_out, int out_size, void* d_ws, size_t ws_size,
                              hipStream_t stream) {
    const float* in  = (const float*)d_in[0];   // [4,64,128,128] fp32
    const float* wgt = (const float*)d_in[1];   // [4,8,25,16384] fp32
    float* out = (float*)d_out;                 // [4,64,128,128] fp32

    dim3 grid(WW / TW, HH / TH, NB);            // (4,16,4) = 256 blocks
    dim3 block(256);                            // 8 wave32s
    localconv_gfx1250_kernel<<<grid, block, TILE_BYTES, stream>>>(in, wgt, out);
}